// ContinuousConvolutionBlock_51187420234486
// MI455X (gfx1250) — compile-verified
//
#include <hip/hip_runtime.h>
#include <hip/hip_bf16.h>

// Problem constants (from reference)
static constexpr int   NN     = 20000;
static constexpr int   C_IN   = 32;
static constexpr int   C_OUT  = 64;
static constexpr int   KF     = 64;       // 4*4*4
static constexpr int   KTOT   = KF * C_IN; // 2048
static constexpr int   EPAD   = 262144;
static constexpr float SCALE  = 20.0f;    // 2 / EXTENT
static constexpr float EPSF   = 1e-12f;
static constexpr float FOUR_OVER_PI = 1.2732395447351628f;

typedef __attribute__((ext_vector_type(2))) float v2f;
typedef __attribute__((ext_vector_type(8))) float v8f;

// ---------------------------------------------------------------- zero scratch
__global__ void zero_f4(float4* __restrict__ p, int n4) {
    int i = blockIdx.x * blockDim.x + threadIdx.x;
    if (i < n4) p[i] = make_float4(0.f, 0.f, 0.f, 0.f);
}

__device__ __forceinline__ float sgnf(float v) {
    return (v > 0.f) ? 1.f : ((v < 0.f) ? -1.f : 0.f);
}

// ------------------------------------------------- edge scatter (wave per edge)
// lane = input channel; 8 corner scatters of 128B coalesced f32 atomics each.
__global__ void edge_scatter(const float* __restrict__ feats,
                             const float* __restrict__ pos,
                             const long long* __restrict__ eq,
                             const long long* __restrict__ en,
                             const float* __restrict__ ev,
                             float* __restrict__ Z) {
    int gid  = blockIdx.x * blockDim.x + threadIdx.x;
    int e    = gid >> 5;
    int lane = gid & 31;
    if (e >= EPAD) return;
    float valid = ev[e];
    if (valid == 0.f) return;                 // padding edges contribute nothing
    int q = (int)eq[e];
    int n = (int)en[e];

    float x = (pos[n * 3 + 0] - pos[q * 3 + 0]) * SCALE;
    float y = (pos[n * 3 + 1] - pos[q * 3 + 1]) * SCALE;
    float z = (pos[n * 3 + 2] - pos[q * 3 + 2]) * SCALE;

    // ball -> cube (volume preserving)
    float sq   = x * x + y * y + z * z;
    float norm = sqrtf(fmaxf(sq, EPSF));
    float rxy2 = x * x + y * y;
    bool  top  = (1.25f * z * z) > rxy2;
    float s_top  = sqrtf(3.0f * norm / (norm + fabsf(z) + EPSF));
    float s_side = norm / sqrtf(fmaxf(rxy2, EPSF));
    float cx = top ? x * s_top : x * s_side;
    float cy = top ? y * s_top : y * s_side;
    float cz = top ? sgnf(z) * norm : 1.5f * z;
    if (sq < EPSF) { cx = 0.f; cy = 0.f; cz = 0.f; }

    float rxy    = sqrtf(fmaxf(cx * cx + cy * cy, EPSF));
    bool  xbig   = fabsf(cy) <= fabsf(cx);
    float safe_x = (fabsf(cx) > EPSF) ? cx : 1.0f;
    float safe_y = (fabsf(cy) > EPSF) ? cy : 1.0f;
    float qx = sgnf(cx) * rxy;
    float qy = sgnf(cy) * rxy;
    float bx = xbig ? qx : qy * FOUR_OVER_PI * atanf(cx / safe_y);
    float by = xbig ? qx * FOUR_OVER_PI * atanf(cy / safe_x) : qy;
    if (cx * cx + cy * cy < EPSF) { bx = 0.f; by = 0.f; }

    // trilinear cell + weights in the 4^3 grid
    float ux = fminf(fmaxf((bx + 1.f) * 0.5f * 3.f, 0.f), 3.f);
    float uy = fminf(fmaxf((by + 1.f) * 0.5f * 3.f, 0.f), 3.f);
    float uz = fminf(fmaxf((cz + 1.f) * 0.5f * 3.f, 0.f), 3.f);
    int i0x = min(max((int)floorf(ux), 0), 2);
    int i0y = min(max((int)floorf(uy), 0), 2);
    int i0z = min(max((int)floorf(uz), 0), 2);
    float fx = ux - (float)i0x;
    float fy = uy - (float)i0y;
    float fz = uz - (float)i0z;

    float fj   = feats[(size_t)n * C_IN + lane] * valid;
    size_t seg = (size_t)q * KF;

    #pragma unroll
    for (int dz = 0; dz < 2; ++dz) {
        float wz = dz ? fz : (1.f - fz);
        #pragma unroll
        for (int dy = 0; dy < 2; ++dy) {
            float wy = dy ? fy : (1.f - fy);
            #pragma unroll
            for (int dx = 0; dx < 2; ++dx) {
                float wx = dx ? fx : (1.f - fx);
                int idx = ((i0z + dz) * 4 + (i0y + dy)) * 4 + (i0x + dx);
                atomicAdd(&Z[(seg + idx) * C_IN + lane], fj * wz * wy * wx);
            }
        }
    }
}

// ------------------------------- conv GEMM: out[n,o] = sum_m Z[n,m] * W[m,o]
// block = 16 nodes, 4 waves; wave w -> output columns [w*16, w*16+16)
// A (16x4 f32): lanes 0-15 rows M, v0/v1 = K,K+1; lanes 16-31 same rows, K+2,K+3
// B (4x16 f32): lane%16 = N col; lane-half selects K / K+2 in v0/v1
__global__ void conv_gemm(const float* __restrict__ Z,   // [NN, 2048]
                          const float* __restrict__ W,   // [2048, 64]
                          float* __restrict__ out) {     // [NN, 64]
    int wave = threadIdx.x >> 5;      // 0..3
    int lane = threadIdx.x & 31;
    int hi   = lane >> 4;             // 0 | 1
    int lo   = lane & 15;
    int nb   = blockIdx.x * 16;

    const float2* arow = (const float2*)(Z + (size_t)(nb + lo) * KTOT + 2 * hi);
    const float*  bcol = W + (size_t)(2 * hi) * C_OUT + wave * 16 + lo;

    v8f c = {};
    #pragma unroll 8
    for (int k = 0; k < KTOT; k += 4) {
        float2 af = arow[k >> 1];
        v2f a; a.x = af.x;                       a.y = af.y;
        v2f b; b.x = bcol[(size_t)k * C_OUT];    b.y = bcol[(size_t)(k + 1) * C_OUT];
        c = __builtin_amdgcn_wmma_f32_16x16x4_f32(false, a, false, b,
                                                  (short)0, c, false, false);
    }
    #pragma unroll
    for (int v = 0; v < 8; ++v)
        out[(size_t)(nb + v + 8 * hi) * C_OUT + wave * 16 + lo] = c[v];
}

// ------------------------------- dense: out[n,o] = feats[n,:] @ Wd + bd[o]
__global__ void dense_gemm(const float* __restrict__ feats, // [NN, 32]
                           const float* __restrict__ Wd,    // [32, 64]
                           const float* __restrict__ bd,    // [64]
                           float* __restrict__ out) {       // [NN, 64]
    int wave = threadIdx.x >> 5;
    int lane = threadIdx.x & 31;
    int hi   = lane >> 4;
    int lo   = lane & 15;
    int nb   = blockIdx.x * 16;
    int o    = wave * 16 + lo;

    const float2* arow = (const float2*)(feats + (size_t)(nb + lo) * C_IN + 2 * hi);
    const float*  bcol = Wd + (size_t)(2 * hi) * C_OUT + o;

    float bias = bd[o];
    v8f c;
    #pragma unroll
    for (int v = 0; v < 8; ++v) c[v] = bias;

    #pragma unroll
    for (int k = 0; k < C_IN; k += 4) {
        float2 af = arow[k >> 1];
        v2f a; a.x = af.x;                    a.y = af.y;
        v2f b; b.x = bcol[k * C_OUT];         b.y = bcol[(k + 1) * C_OUT];
        c = __builtin_amdgcn_wmma_f32_16x16x4_f32(false, a, false, b,
                                                  (short)0, c, false, false);
    }
    #pragma unroll
    for (int v = 0; v < 8; ++v)
        out[(size_t)(nb + v + 8 * hi) * C_OUT + o] = c[v];
}

// ----------------------------------------------------------------------------
extern "C" void kernel_launch(void* const* d_in, const int* in_sizes, int n_in,
                              void* d_out, int out_size, void* d_ws, size_t ws_size,
                              hipStream_t stream) {
    const float*     feats = (const float*)d_in[0];
    const float*     pos   = (const float*)d_in[1];
    const float*     Wconv = (const float*)d_in[2];   // [64,32,64] flat == [2048,64]
    const float*     Wd    = (const float*)d_in[3];
    const float*     bd    = (const float*)d_in[4];
    const long long* eq    = (const long long*)d_in[5];
    const long long* en    = (const long long*)d_in[6];
    const float*     ev    = (const float*)d_in[7];

    float* out_conv  = (float*)d_out;                         // [NN, 64]
    float* out_dense = out_conv + (size_t)NN * C_OUT;         // [NN, 64]
    float* Z         = (float*)d_ws;                          // [NN*KF, 32] = 164 MB

    // 1) zero the segment-sum scratch (must be redone every call)
    int n4 = NN * KF * C_IN / 4;   // 10.24M float4
    zero_f4<<<(n4 + 255) / 256, 256, 0, stream>>>((float4*)Z, n4);

    // 2) edge scatter: one wave per edge, lane = channel
    edge_scatter<<<EPAD * 32 / 256, 256, 0, stream>>>(feats, pos, eq, en, ev, Z);

    // 3) conv contraction on the matrix pipe (f32 WMMA, exact)
    conv_gemm<<<NN / 16, 128, 0, stream>>>(Z, Wconv, out_conv);

    // 4) dense head (f32 WMMA, bias in C)
    dense_gemm<<<NN / 16, 128, 0, stream>>>(feats, Wd, bd, out_dense);
}